// Quantizer_26096221290929
// MI455X (gfx1250) — compile-verified
//
#include <hip/hip_runtime.h>
#include <hip/hip_bf16.h>

typedef float v2f __attribute__((ext_vector_type(2)));
typedef float v8f __attribute__((ext_vector_type(8)));

#define NROWS   32768      // B*H*W
#define KCODES  8192
#define DIM     64
#define CHUNK   128        // codebook rows staged in LDS per iteration
#define LPITCH  68         // LDS floats per codebook row (bank-conflict-free)
#define QELEMS  2097152    // 8*64*64*64

// --- kernel 1: codebook norms + zero loss accumulators -----------------------
__global__ __launch_bounds__(256) void vq_prep(const float* __restrict__ emb,
                                               float* __restrict__ norms,
                                               float* __restrict__ loss_c,
                                               float* __restrict__ loss_b) {
    int k = blockIdx.x * blockDim.x + threadIdx.x;       // 0..8191
    if (k == 0) { *loss_c = 0.0f; *loss_b = 0.0f; }
    const float4* e = (const float4*)(emb + (size_t)k * DIM);
    float s = 0.0f;
#pragma unroll
    for (int q = 0; q < DIM / 4; ++q) {
        float4 v = e[q];
        s += v.x * v.x + v.y * v.y + v.z * v.z + v.w * v.w;
    }
    norms[k] = s;
}

// --- kernel 2: WMMA distance sweep + argmin ---------------------------------
// Block = 128 threads = 4 waves; each wave owns 16 x-rows, all waves share the
// LDS-staged codebook chunk. Distances: d'(n,k) = ||e_k||^2 - 2 * <x_n, e_k>.
__global__ __launch_bounds__(128) void vq_argmin(const float* __restrict__ x,
                                                 const float* __restrict__ emb,
                                                 const float* __restrict__ norms,
                                                 int*   __restrict__ idx_i,
                                                 float* __restrict__ idx_f) {
    __shared__ float lds_e[CHUNK * LPITCH];
    __shared__ float lds_n[CHUNK];

    const int lane = threadIdx.x & 31;
    const int wave = threadIdx.x >> 5;
    const int nlo  = lane & 15;          // A row / B column within tile
    const int h    = lane >> 4;          // K-half select (dims 2h, 2h+1 of each 4)

    const int row_base = blockIdx.x * 64 + wave * 16;
    const int m  = row_base + nlo;       // this lane's x-row (0..32767)
    const int b  = m >> 12;              // batch
    const int hw = m & 4095;             // h*64+w
    const float* xrow = x + (size_t)b * 262144 + hw;   // + d*4096 selects channel d

    // A-matrix (16x4 f32 layout): a[s] = { x[m][4s+2h], x[m][4s+2h+1] }, s=0..15
    v2f a[16];
#pragma unroll
    for (int s = 0; s < 16; ++s) {
        int d0 = 4 * s + 2 * h;
        a[s].x = xrow[(size_t)d0 * 4096];
        a[s].y = xrow[(size_t)(d0 + 1) * 4096];
    }

    float best[8];
    int   bidx[8];
#pragma unroll
    for (int r = 0; r < 8; ++r) { best[r] = 3.4e38f; bidx[r] = 0; }

    for (int chunk = 0; chunk < KCODES; chunk += CHUNK) {
        // cooperative stage: one codebook row per thread
        {
            const float* src = emb + (size_t)(chunk + threadIdx.x) * DIM;
            float* dst = lds_e + threadIdx.x * LPITCH;
#pragma unroll
            for (int q = 0; q < DIM / 4; ++q)
                *(float4*)(dst + 4 * q) = ((const float4*)src)[q];
            lds_n[threadIdx.x] = norms[chunk + threadIdx.x];
        }
        __syncthreads();

        for (int t = 0; t < CHUNK / 16; ++t) {          // 16-code tiles
            const int col_l = t * 16 + nlo;
            const int col_g = chunk + col_l;
            const float nrm = lds_n[col_l];
            const float* erow = lds_e + col_l * LPITCH;

            v8f acc = {};
#pragma unroll
            for (int s = 0; s < 16; ++s) {
                // B-matrix (4x16 f32): column n=lane&15, K = 2h,2h+1 of dims 4s..4s+3
                v2f bb = *(const v2f*)(erow + 4 * s + 2 * h);
                acc = __builtin_amdgcn_wmma_f32_16x16x4_f32(
                        false, a[s], false, bb, (short)0, acc, false, false);
            }
#pragma unroll
            for (int r = 0; r < 8; ++r) {
                float d = nrm - 2.0f * acc[r];
                if (d < best[r]) { best[r] = d; bidx[r] = col_g; }
            }
        }
        __syncthreads();
    }

    // cross-lane argmin within each 16-lane half (half h holds rows r + 8h)
#pragma unroll
    for (int r = 0; r < 8; ++r) {
        float v = best[r];
        int   i = bidx[r];
#pragma unroll
        for (int msk = 8; msk >= 1; msk >>= 1) {
            float ov = __shfl_xor(v, msk, 32);
            int   oi = __shfl_xor(i, msk, 32);
            if (ov < v || (ov == v && oi < i)) { v = ov; i = oi; }
        }
        if (nlo == 0) {
            int row = row_base + r + h * 8;
            idx_i[row] = i;
            idx_f[row] = (float)i;
        }
    }
}

// --- kernel 3: gather quantized output + loss reduction ---------------------
// gid enumerates output in native (B, C, H, W) order, which for C==H==W==64 is
// exactly what the reference's reshape/transpose chain produces, and is also
// x's native layout -> both load and store are coalesced.
__global__ __launch_bounds__(256) void vq_scatter(const float* __restrict__ x,
                                                  const float* __restrict__ emb,
                                                  const int*   __restrict__ idx_i,
                                                  float* __restrict__ out,
                                                  float* __restrict__ loss_c,
                                                  float* __restrict__ loss_b) {
    int gid = blockIdx.x * blockDim.x + threadIdx.x;     // 0..QELEMS-1
    int b   = gid >> 18;
    int c   = (gid >> 12) & 63;
    int hw  = gid & 4095;
    int n   = (b << 12) | hw;

    int   k  = idx_i[n];
    float q  = emb[(size_t)k * DIM + c];
    float xr = x[gid];
    out[gid] = q;

    float d = q - xr;
    float e = d * d * (1.0f / (float)QELEMS);
#pragma unroll
    for (int msk = 16; msk >= 1; msk >>= 1)
        e += __shfl_xor(e, msk, 32);
    if ((threadIdx.x & 31) == 0) {
        atomicAdd(loss_c, e);
        atomicAdd(loss_b, e);
    }
}

extern "C" void kernel_launch(void* const* d_in, const int* in_sizes, int n_in,
                              void* d_out, int out_size, void* d_ws, size_t ws_size,
                              hipStream_t stream) {
    const float* x   = (const float*)d_in[0];   // (8,64,64,64) fp32
    const float* emb = (const float*)d_in[1];   // (8192,64)    fp32

    float* out    = (float*)d_out;
    float* loss_c = out + QELEMS;               // commitment_loss
    float* loss_b = out + QELEMS + 1;           // codebook_loss
    float* idx_f  = out + QELEMS + 2;           // indices (as float)

    float* norms = (float*)d_ws;                                // 8192 f32
    int*   idx_i = (int*)((char*)d_ws + KCODES * sizeof(float)); // 32768 i32

    vq_prep   <<<KCODES / 256, 256, 0, stream>>>(emb, norms, loss_c, loss_b);
    vq_argmin <<<NROWS / 64, 128, 0, stream>>>(x, emb, norms, idx_i, idx_f);
    vq_scatter<<<QELEMS / 256, 256, 0, stream>>>(x, emb, idx_i, out, loss_c, loss_b);
}